// GCN_39427799777294
// MI455X (gfx1250) — compile-verified
//
#include <hip/hip_runtime.h>
#include <hip/hip_bf16.h>

typedef __attribute__((ext_vector_type(2))) float v2f;
typedef __attribute__((ext_vector_type(8))) float v8f;

#define NF 512
#define NH 64
#define NC 8
#define KTILE 128
#define PAIRS (KTILE / 2)   // 64 k-pairs per LDS chunk
#define PSTR 160            // words per k-pair row: 128 data + 32 pad (pair p+1 lands 32 banks away)

// ---------------- degree / normalization ----------------

__global__ void deg_init_kernel(float* deg, int n) {
  int i = blockIdx.x * blockDim.x + threadIdx.x;
  if (i < n) deg[i] = 1.0f;   // self-loop contribution
}

__global__ void deg_count_kernel(const int* __restrict__ dst, float* deg, int e) {
  int i = blockIdx.x * blockDim.x + threadIdx.x;
  if (i < e) atomicAdd(&deg[dst[i]], 1.0f);
}

__global__ void dinv_kernel(float* deg, int n) {
  int i = blockIdx.x * blockDim.x + threadIdx.x;
  if (i < n) deg[i] = rsqrtf(deg[i]);  // deg >= 1 always
}

// ---------------- layer 1: h_scaled = (X @ W1) * dinv[row], acc1 = h_scaled ----------------

__global__ __launch_bounds__(256) void gemm1_wmma_kernel(
    const float* __restrict__ X, const float* __restrict__ W1,
    const float* __restrict__ dinv, float* __restrict__ hs,
    float* __restrict__ acc1, int n) {
  __shared__ float wl[PAIRS * PSTR];          // 40 KB, K-pair-interleaved W1 chunk
  const int lane  = threadIdx.x & 31;
  const int wave  = threadIdx.x >> 5;
  const int m0    = blockIdx.x * 128 + wave * 16;
  const int mlan  = lane & 15;
  const int khalf = (lane >> 4) << 1;         // 0 for lanes 0-15, 2 for lanes 16-31
  const int pbase = lane >> 4;                // extra k-pair for upper half-wave

  int row = m0 + mlan;
  if (row >= n) row = n - 1;                  // clamp: keep EXEC all-ones for WMMA
  const float* xrow = X + (size_t)row * NF;

  v8f c0 = {}; v8f c1 = {}; v8f c2 = {}; v8f c3 = {};

  for (int kb = 0; kb < NF; kb += KTILE) {
    // stage W1[kb..kb+128) x 64 into LDS, element (k,n) -> word (k/2)*PSTR + 2n + (k&1)
    for (int i = threadIdx.x; i < KTILE * NH; i += 256) {
      int kl = i >> 6, nn = i & 63;
      wl[(kl >> 1) * PSTR + (nn << 1) + (kl & 1)] = W1[(size_t)kb * NH + i];
    }
    if (kb + KTILE < NF)
      __builtin_prefetch(xrow + kb + KTILE, 0, 3);   // stream next X chunk toward L2/L0
    __syncthreads();

    for (int kk = 0; kk < KTILE; kk += 4) {
      v2f a; a.x = xrow[kb + kk + khalf]; a.y = xrow[kb + kk + khalf + 1];  // A: 16x4 f32 frag
      const v2f* wp = (const v2f*)(wl + ((kk >> 1) + pbase) * PSTR);        // B pairs, b64 loads
      v2f b0 = wp[mlan];
      v2f b1 = wp[16 + mlan];
      v2f b2 = wp[32 + mlan];
      v2f b3 = wp[48 + mlan];
      c0 = __builtin_amdgcn_wmma_f32_16x16x4_f32(false, a, false, b0, (short)0, c0, false, false);
      c1 = __builtin_amdgcn_wmma_f32_16x16x4_f32(false, a, false, b1, (short)0, c1, false, false);
      c2 = __builtin_amdgcn_wmma_f32_16x16x4_f32(false, a, false, b2, (short)0, c2, false, false);
      c3 = __builtin_amdgcn_wmma_f32_16x16x4_f32(false, a, false, b3, (short)0, c3, false, false);
    }
    __syncthreads();
  }

  // C layout: VGPR r, lanes 0-15 -> M=r, lanes 16-31 -> M=r+8; N = lane%16 (+16*tile)
  const int half8 = (lane >> 4) << 3;
  for (int r = 0; r < 8; ++r) {
    int rr = m0 + r + half8;
    if (rr < n) {
      float dv = dinv[rr];
      size_t base = (size_t)rr * NH + mlan;
      float v0 = c0[r] * dv, v1 = c1[r] * dv, v2 = c2[r] * dv, v3 = c3[r] * dv;
      hs[base]      = v0; hs[base + 16]   = v1; hs[base + 32]   = v2; hs[base + 48]   = v3;
      acc1[base]    = v0; acc1[base + 16] = v1; acc1[base + 32] = v2; acc1[base + 48] = v3;
    }
  }
}

// acc1[dst] += h_scaled[src] over all edges (self-loop already in acc1 init); 2 feats/thread
__global__ void agg1_kernel(const int* __restrict__ src, const int* __restrict__ dst,
                            const float* __restrict__ hs, float* acc1, long long total) {
  long long idx = (long long)blockIdx.x * blockDim.x + threadIdx.x;
  if (idx >= total) return;
  int e = (int)(idx >> 5);
  int f = (int)(idx & 31) << 1;
  int s = src[e], d = dst[e];
  const float2 v = *(const float2*)(hs + (size_t)s * NH + f);
  float* o = acc1 + (size_t)d * NH + f;
  atomicAdd(o, v.x);
  atomicAdd(o + 1, v.y);
}

// ---------------- layer 2: t2s = dinv * (relu(acc1*dinv + b1) @ W2), acc2 = t2s ----------------

__global__ void gemm2_kernel(const float* __restrict__ acc1, const float* __restrict__ dinv,
                             const float* __restrict__ b1, const float* __restrict__ W2,
                             float* __restrict__ t2s, float* __restrict__ acc2, int n) {
  int idx = blockIdx.x * blockDim.x + threadIdx.x;
  if (idx >= n * NC) return;
  int i = idx >> 3, c = idx & 7;
  float di = dinv[i];
  const float* arow = acc1 + (size_t)i * NH;
  float s = 0.0f;
  for (int k = 0; k < NH; ++k) {
    float h = fmaxf(fmaf(arow[k], di, b1[k]), 0.0f);   // finalize layer-1 + ReLU, fused
    s = fmaf(h, W2[k * NC + c], s);
  }
  float v = s * di;
  t2s[idx]  = v;
  acc2[idx] = v;
}

__global__ void agg2_kernel(const int* __restrict__ src, const int* __restrict__ dst,
                            const float* __restrict__ t2s, float* acc2, long long total) {
  long long idx = (long long)blockIdx.x * blockDim.x + threadIdx.x;
  if (idx >= total) return;
  int e = (int)(idx >> 2);
  int c = (int)(idx & 3) << 1;
  int s = src[e], d = dst[e];
  const float2 v = *(const float2*)(t2s + (size_t)s * NC + c);
  float* o = acc2 + (size_t)d * NC + c;
  atomicAdd(o, v.x);
  atomicAdd(o + 1, v.y);
}

__global__ void finalize2_kernel(const float* __restrict__ acc2, const float* __restrict__ dinv,
                                 const float* __restrict__ b2, float* __restrict__ out, int n) {
  int idx = blockIdx.x * blockDim.x + threadIdx.x;
  if (idx >= n * NC) return;
  int i = idx >> 3, c = idx & 7;
  out[idx] = fmaf(acc2[idx], dinv[i], b2[c]);
}

// ---------------- launcher ----------------

extern "C" void kernel_launch(void* const* d_in, const int* in_sizes, int n_in,
                              void* d_out, int out_size, void* d_ws, size_t ws_size,
                              hipStream_t stream) {
  const float* x  = (const float*)d_in[0];
  const int*   ei = (const int*)d_in[1];
  const float* W1 = (const float*)d_in[2];
  const float* b1 = (const float*)d_in[3];
  const float* W2 = (const float*)d_in[4];
  const float* b2 = (const float*)d_in[5];
  float* out = (float*)d_out;

  const int n = in_sizes[0] / NF;
  const int e = in_sizes[1] / 2;
  const int* src = ei;        // edge_index[0]
  const int* dst = ei + e;    // edge_index[1]

  float* ws   = (float*)d_ws;
  float* dinv = ws;                                 // n        (deg in-place -> dinv)
  float* hs   = dinv + n;                           // n*64
  float* acc1 = hs + (size_t)n * NH;                // n*64
  float* t2s  = acc1 + (size_t)n * NH;              // n*8
  float* acc2 = t2s + (size_t)n * NC;               // n*8

  const int T = 256;
  deg_init_kernel<<<(n + T - 1) / T, T, 0, stream>>>(dinv, n);
  deg_count_kernel<<<(e + T - 1) / T, T, 0, stream>>>(dst, dinv, e);
  dinv_kernel<<<(n + T - 1) / T, T, 0, stream>>>(dinv, n);

  gemm1_wmma_kernel<<<(n + 127) / 128, T, 0, stream>>>(x, W1, dinv, hs, acc1, n);

  long long tot1 = (long long)e * 32;   // 2 features per thread
  agg1_kernel<<<(unsigned)((tot1 + T - 1) / T), T, 0, stream>>>(src, dst, hs, acc1, tot1);

  gemm2_kernel<<<(n * NC + T - 1) / T, T, 0, stream>>>(acc1, dinv, b1, W2, t2s, acc2, n);

  long long tot2 = (long long)e * 4;    // 2 classes per thread
  agg2_kernel<<<(unsigned)((tot2 + T - 1) / T), T, 0, stream>>>(src, dst, t2s, acc2, tot2);

  finalize2_kernel<<<(n * NC + T - 1) / T, T, 0, stream>>>(acc2, dinv, b2, out, n);
}